// SR_GNN_19524921328095
// MI455X (gfx1250) — compile-verified
//
#include <hip/hip_runtime.h>
#include <hip/hip_bf16.h>

typedef float        v2f __attribute__((ext_vector_type(2)));
typedef float        v8f __attribute__((ext_vector_type(8)));
typedef unsigned int u32;
typedef u32          v4u __attribute__((ext_vector_type(4)));
typedef int          v4i __attribute__((ext_vector_type(4)));
typedef int          v8i __attribute__((ext_vector_type(8)));

#define WMMA_F32(a, b, c) \
  __builtin_amdgcn_wmma_f32_16x16x4_f32(false, (a), false, (b), (short)0, (c), false, false)

static constexpr int B_  = 512;
static constexpr int N_  = 200;
static constexpr int D_  = 128;
static constexpr int M_  = B_ * N_;   // 102400
static constexpr int NC1 = 640;       // [hv_in(128) | hv_out(128) | gh(384)]
static constexpr int NC2 = 384;       // gi columns

// ---- LDS byte offset of a __shared__ object (low 32 bits of flat addr) ----
__device__ __forceinline__ u32 lds_off(const void* p) {
  return (u32)(size_t)p;
}

// ---- Tensor Data Mover: DMA a 2-D fp32 tile (tile_w x tile_h elems, row
//      stride `stride` elems in global) into contiguous LDS at lds_addr. ----
__device__ __forceinline__ void tdm_load_2d(const void* gptr, u32 lds_addr,
                                            u32 tile_w, u32 tile_h, u32 stride) {
  unsigned long long ga = (unsigned long long)(size_t)gptr;
  // D# group 0: count=1 | lds_addr | global_addr[56:0] | type=2
  v4u g0 = { 1u, lds_addr, (u32)ga,
             (u32)((ga >> 32) & 0x01FFFFFFull) | (2u << 30) };
  // D# group 1: data_size=2 (4B); tensor_dim0/1 = tile (exact, no OOB);
  //             tile_dim0/1; tensor_dim0_stride = row stride.
  v8i g1 = { (int)(2u << 16),
             (int)((tile_w & 0xFFFFu) << 16),
             (int)(((tile_w >> 16) & 0xFFFFu) | ((tile_h & 0xFFFFu) << 16)),
             (int)(((tile_h >> 16) & 0xFFFFu) | ((tile_w & 0xFFFFu) << 16)),
             (int)(tile_h & 0xFFFFu),
             (int)stride,
             0, 0 };
  v4i z4 = { 0, 0, 0, 0 };
#if __clang_major__ >= 23
  v8i z8 = { 0, 0, 0, 0, 0, 0, 0, 0 };
  __builtin_amdgcn_tensor_load_to_lds(g0, g1, z4, z4, z8, 0);
#else
  __builtin_amdgcn_tensor_load_to_lds(g0, g1, z4, z4, 0);
#endif
}

// ---------------- K1: hidden = emb[x], float4 vectorized ----------------
__global__ __launch_bounds__(256) void k_gather(const int* __restrict__ x,
                                                const float* __restrict__ emb,
                                                float* __restrict__ hidden) {
  int i = blockIdx.x * 256 + threadIdx.x;   // float4 index, 3,276,800 total (exact)
  int m  = i >> 5;
  int c4 = i & 31;
  long tok = x[m];
  float4 v = *(const float4*)(emb + (size_t)tok * D_ + c4 * 4);
  *(float4*)(hidden + (size_t)m * D_ + c4 * 4) = v;
}

// ---------------- K0: transposed weight panels + concat bias ----------------
__global__ __launch_bounds__(256) void k_prep(const float* __restrict__ w_in,
                                              const float* __restrict__ w_out,
                                              const float* __restrict__ w_hh,
                                              const float* __restrict__ b_in,
                                              const float* __restrict__ b_out,
                                              const float* __restrict__ b_hh,
                                              const float* __restrict__ w_ih,
                                              float* __restrict__ wcat,
                                              float* __restrict__ bcat,
                                              float* __restrict__ wih_t) {
  int i = blockIdx.x * 256 + threadIdx.x;
  if (i < 128 * 640) {
    int k = i / 640, j = i % 640;
    float v;
    if (j < 128)       v = w_in[j * 128 + k];
    else if (j < 256)  v = w_out[(j - 128) * 128 + k];
    else               v = w_hh[(j - 256) * 128 + k];
    wcat[i] = v;
  } else if (i < 128 * 640 + 256 * 384) {
    int t = i - 128 * 640;
    int k = t / 384, j = t % 384;
    wih_t[t] = w_ih[j * 256 + k];
  } else if (i < 128 * 640 + 256 * 384 + 640) {
    int j = i - (128 * 640 + 256 * 384);
    bcat[j] = (j < 128) ? b_in[j] : (j < 256) ? b_out[j - 128] : b_hh[j - 256];
  }
}

// ---------------- Generic GEMM: Yo = X(MxK) * W(KxNOUT) + bias ----------------
// Block: 8 waves, tile 128(M) x 64(N); wave: 16(M) x 64(N) -> 4 accumulators.
// W staged in LDS by the Tensor Data Mover, double-buffered in 64-K chunks so
// the DMA for chunk c+1 overlaps the WMMA work on chunk c. A fragments for a
// whole chunk are prefetched into registers before the LDS handoff.
template <int K, int NOUT>
__global__ __launch_bounds__(256) void k_gemm_rw(const float* __restrict__ X,
                                                 const float* __restrict__ W,
                                                 const float* __restrict__ bias,
                                                 float* __restrict__ Yo) {
  constexpr int KC     = 64;
  constexpr int NCHUNK = K / KC;
  constexpr int NFRAG  = KC / 4;              // 16 A fragments per chunk
  __shared__ float wlds[2][KC * 64];          // 2 x 16 KB
  const int tid   = threadIdx.x;
  const int nBase = blockIdx.y * 64;
  const int mBase = blockIdx.x * 128;
  const int wave  = tid >> 5;
  const int lane  = tid & 31;
  const int mrow  = mBase + wave * 16 + (lane & 15);
  const int koff  = (lane >> 4) << 1;         // 0 or 2
  const int ncol  = lane & 15;
  const bool issuer = (wave == 0);

  v8f acc[4] = {};
  const float* xrow = X + (size_t)mrow * K;

  if (issuer)
    tdm_load_2d(W + nBase, lds_off(&wlds[0][0]), 64, KC, NOUT);

#pragma unroll
  for (int c = 0; c < NCHUNK; ++c) {
    // Prefetch this chunk's A fragments; loads fly during barrier + TDM wait.
    v2f areg[NFRAG];
#pragma unroll
    for (int j = 0; j < NFRAG; ++j)
      areg[j] = *(const v2f*)(xrow + c * KC + j * 4 + koff);

    __syncthreads();                          // buf[(c+1)&1] free to overwrite
    if (c + 1 < NCHUNK) {
      if (issuer) {
        tdm_load_2d(W + (size_t)(c + 1) * KC * NOUT + nBase,
                    lds_off(&wlds[(c + 1) & 1][0]), 64, KC, NOUT);
        __builtin_amdgcn_s_wait_tensorcnt(1); // chunk c complete (in-order)
      }
    } else if (issuer) {
      __builtin_amdgcn_s_wait_tensorcnt(0);
    }
    __syncthreads();
    const float* bufb = &wlds[c & 1][0];
#pragma unroll
    for (int j = 0; j < NFRAG; ++j) {
      const float* bb = bufb + (j * 4 + koff) * 64 + ncol;
#pragma unroll
      for (int t = 0; t < 4; ++t) {
        v2f b;
        b.x = bb[t * 16];
        b.y = bb[t * 16 + 64];
        acc[t] = WMMA_F32(areg[j], b, acc[t]);
      }
    }
  }

#pragma unroll
  for (int t = 0; t < 4; ++t) {
    int col = nBase + t * 16 + ncol;
    float bv = bias[col];
#pragma unroll
    for (int r = 0; r < 8; ++r) {
      int m = mBase + wave * 16 + r + ((lane >> 4) << 3);
      Yo[(size_t)m * NOUT + col] = acc[t][r] + bv;
    }
  }
}

// ---------------- K3: batched adjacency matmul ----------------
// inputs[b, m, h*128 + c] = sum_k A[b, m, h*200 + k] * hv_h[b, k, c] + b_?ah[c]
// hv panel (200 x 128 slice of Y, row stride 640) TDM-staged in 40-row
// double-buffered chunks. Block: 8 waves = 4(M) x 2(N), tile 64(M) x 128(N).
// Out-of-range M rows are CLAMPED (not predicated): their loads read valid
// memory and their accumulator rows are simply never stored.
__global__ __launch_bounds__(256) void k_adj(const float* __restrict__ A,
                                             const float* __restrict__ Y,
                                             const float* __restrict__ b_iah,
                                             const float* __restrict__ b_oah,
                                             float* __restrict__ inputs) {
  constexpr int KC    = 40;
  constexpr int NCH   = 5;                    // 5 * 40 = 200
  constexpr int NFRAG = KC / 4;               // 10 A fragments per chunk
  __shared__ float hv[2][KC * 128];           // 2 x 20 KB
  const int tid   = threadIdx.x;
  const int b     = blockIdx.z;
  const int half  = blockIdx.y;
  const int mBase = blockIdx.x * 64;
  const int wave  = tid >> 5;
  const int lane  = tid & 31;
  const int wm = wave & 3;                    // 4 waves in M
  const int wn = wave >> 2;                   // 2 waves in N
  const int koff = (lane >> 4) << 1;
  const int ncol = lane & 15;
  const int rowL = mBase + wm * 16 + (lane & 15);
  const int rowC = (rowL < N_) ? rowL : (N_ - 1);   // clamped, always in-bounds
  const bool issuer = (wave == 0);
  const int coff = half * 128;

  const float* arow = A + ((size_t)b * N_ + rowC) * 400 + half * 200;
  const float* Yp   = Y + (size_t)b * N_ * NC1 + coff;   // row stride 640

  v8f acc[4] = {};

  if (issuer)
    tdm_load_2d(Yp, lds_off(&hv[0][0]), 128, KC, NC1);

#pragma unroll
  for (int c = 0; c < NCH; ++c) {
    v2f areg[NFRAG];
#pragma unroll
    for (int j = 0; j < NFRAG; ++j)
      areg[j] = *(const v2f*)(arow + c * KC + j * 4 + koff);

    __syncthreads();
    if (c + 1 < NCH) {
      if (issuer) {
        tdm_load_2d(Yp + (size_t)(c + 1) * KC * NC1,
                    lds_off(&hv[(c + 1) & 1][0]), 128, KC, NC1);
        __builtin_amdgcn_s_wait_tensorcnt(1);
      }
    } else if (issuer) {
      __builtin_amdgcn_s_wait_tensorcnt(0);
    }
    __syncthreads();
    const float* bufb = &hv[c & 1][0];
#pragma unroll
    for (int j = 0; j < NFRAG; ++j) {
      const float* bb = bufb + (j * 4 + koff) * 128 + wn * 64 + ncol;
#pragma unroll
      for (int t = 0; t < 4; ++t) {
        v2f bv;
        bv.x = bb[t * 16];
        bv.y = bb[t * 16 + 128];
        acc[t] = WMMA_F32(areg[j], bv, acc[t]);
      }
    }
  }

  const float* bah = half ? b_oah : b_iah;
#pragma unroll
  for (int t = 0; t < 4; ++t) {
    int col = wn * 64 + t * 16 + ncol;        // 0..127
    float bv = bah[col];
#pragma unroll
    for (int r = 0; r < 8; ++r) {
      int m = mBase + wm * 16 + r + ((lane >> 4) << 3);
      if (m < N_)
        inputs[((size_t)b * N_ + m) * 256 + coff + col] = acc[t][r] + bv;
    }
  }
}

// ---------------- K5: fused GRU gates ----------------
__device__ inline float gru1(float ir, float ii, float inn,
                             float hr, float hi, float hn, float h) {
  float g_i = 1.0f / (1.0f + __expf(-(ii + hi)));
  float g_r = 1.0f / (1.0f + __expf(-(ir + hr)));
  float g_o = tanhf(inn + g_r * hn);
  return g_o + g_i * (h - g_o);
}

__global__ __launch_bounds__(256) void k_gate(const float* __restrict__ gi,
                                              const float* __restrict__ Y,
                                              const float* __restrict__ hidden,
                                              float* __restrict__ out) {
  int i = blockIdx.x * 256 + threadIdx.x;     // float4 index over (M_, 128)
  size_t m = (size_t)(i >> 5);
  int d = (i & 31) * 4;
  const float* gim = gi + m * NC2;
  const float* ghm = Y + m * NC1 + 256;       // gh = Y[:, 256:640]
  float4 ir  = *(const float4*)(gim + d);
  float4 ii  = *(const float4*)(gim + 128 + d);
  float4 inn = *(const float4*)(gim + 256 + d);
  float4 hr  = *(const float4*)(ghm + d);
  float4 hi  = *(const float4*)(ghm + 128 + d);
  float4 hn  = *(const float4*)(ghm + 256 + d);
  float4 hd  = *(const float4*)(hidden + m * D_ + d);
  float4 o;
  o.x = gru1(ir.x, ii.x, inn.x, hr.x, hi.x, hn.x, hd.x);
  o.y = gru1(ir.y, ii.y, inn.y, hr.y, hi.y, hn.y, hd.y);
  o.z = gru1(ir.z, ii.z, inn.z, hr.z, hi.z, hn.z, hd.z);
  o.w = gru1(ir.w, ii.w, inn.w, hr.w, hi.w, hn.w, hd.w);
  *(float4*)(out + m * D_ + d) = o;
}

// ---------------- host launch ----------------
extern "C" void kernel_launch(void* const* d_in, const int* in_sizes, int n_in,
                              void* d_out, int out_size, void* d_ws, size_t ws_size,
                              hipStream_t stream) {
  const int*   x     = (const int*)  d_in[0];
  const float* A     = (const float*)d_in[1];
  const float* emb   = (const float*)d_in[2];
  const float* w_in  = (const float*)d_in[3];
  const float* b_in  = (const float*)d_in[4];
  const float* w_out = (const float*)d_in[5];
  const float* b_out = (const float*)d_in[6];
  const float* w_ih  = (const float*)d_in[7];
  const float* b_ih  = (const float*)d_in[8];
  const float* w_hh  = (const float*)d_in[9];
  const float* b_hh  = (const float*)d_in[10];
  const float* b_iah = (const float*)d_in[11];
  const float* b_oah = (const float*)d_in[12];
  float* out = (float*)d_out;

  float* ws     = (float*)d_ws;
  float* hidden = ws;                               // M_ * 128
  float* Y      = hidden + (size_t)M_ * D_;         // M_ * 640  [hv_in|hv_out|gh]
  float* inputs = Y + (size_t)M_ * NC1;             // M_ * 256
  float* gi     = inputs + (size_t)M_ * 256;        // M_ * 384
  float* wcat   = gi + (size_t)M_ * NC2;            // 128*640
  float* wih_t  = wcat + 128 * 640;                 // 256*384
  float* bcat   = wih_t + 256 * 384;                // 640

  // 1) gather
  k_gather<<<(M_ * D_ / 4) / 256, 256, 0, stream>>>(x, emb, hidden);
  // 2) weight prep
  int prepN = 128 * 640 + 256 * 384 + 640;
  k_prep<<<(prepN + 255) / 256, 256, 0, stream>>>(w_in, w_out, w_hh, b_in, b_out,
                                                  b_hh, w_ih, wcat, bcat, wih_t);
  // 3) Y = hidden @ Wcat + bcat   (M=102400, K=128, N=640)
  k_gemm_rw<128, NC1><<<dim3(M_ / 128, NC1 / 64), 256, 0, stream>>>(hidden, wcat, bcat, Y);
  // 4) inputs = [A_in @ hv_in + b_iah | A_out @ hv_out + b_oah]
  k_adj<<<dim3(4, 2, B_), 256, 0, stream>>>(A, Y, b_iah, b_oah, inputs);
  // 5) gi = inputs @ w_ih^T + b_ih  (M=102400, K=256, N=384)
  k_gemm_rw<256, NC2><<<dim3(M_ / 128, NC2 / 64), 256, 0, stream>>>(inputs, wih_t, b_ih, gi);
  // 6) GRU gates
  k_gate<<<(M_ * D_ / 4) / 256, 256, 0, stream>>>(gi, Y, hidden, out);
}